// GRU_66314295050287
// MI455X (gfx1250) — compile-verified
//
#include <hip/hip_runtime.h>

typedef __attribute__((ext_vector_type(2))) float v2f;
typedef __attribute__((ext_vector_type(8))) float v8f;
typedef __attribute__((ext_vector_type(4))) unsigned int u32x4;
typedef __attribute__((ext_vector_type(8))) int i32x8;
typedef __attribute__((ext_vector_type(4))) int i32x4;

#define kT        12
#define kFE       32
#define kSEQ      16                    // sequences per tile (B-matrix columns)
#define kGRPS     4                     // tiles per wave (double-buffered pipeline)
#define kSEQ_DW   (kFE * kT)            // 384 dwords per sequence
#define kROWSTR   387                   // 384 + 3 pad dwords (TDM pads 1 dw / 128 dw)
#define kBUF_DW   6208                  // 6144 data + 47 pads + slack, 16B aligned

// One TDM descriptor: contiguous 16x384 f32 tile, global -> LDS, with
// 1-DWORD padding inserted every 128 DWORDs (-> row stride 387 in LDS).
__device__ __forceinline__ void tdm_load_tile(const float* gsrc, unsigned lds_byte)
{
    uint64_t ga = (uint64_t)(uintptr_t)gsrc;
    u32x4 d0;
    d0[0] = 1u;                                                  // count=1 (valid), user mode
    d0[1] = lds_byte;                                            // lds_addr
    d0[2] = (unsigned)ga;                                        // global_addr[31:0]
    d0[3] = ((unsigned)(ga >> 32) & 0x01FFFFFFu) | (2u << 30);   // global_addr[56:32] | type=2

    i32x8 d1;
    d1[0] = (2 << 16) | (1 << 20) | (6 << 22);  // data_size=4B | pad_enable | pad_interval=128dw (pad_amount=1dw)
    d1[1] = (kSEQ_DW << 16);                    // tensor_dim0 = 384 (lo16; hi bits land in d1[2][15:0]=0)
    d1[2] = (kSEQ << 16);                       // tensor_dim1 = 16
    d1[3] = (kSEQ_DW << 16);                    // tile_dim0 = 384
    d1[4] = kSEQ;                               // tile_dim1 = 16, tile_dim2 = 0
    d1[5] = kSEQ_DW;                            // tensor_dim0_stride = 384 (lo32)
    d1[6] = ((kSEQ_DW * kSEQ) << 16);           // dim0_stride hi16=0 | tensor_dim1_stride lo16 = 6144
    d1[7] = 0;                                  // tensor_dim1_stride hi32 = 0

    i32x4 dz4 = {};                             // groups 2/3 unused (2-D tensor)
    i32x8 dz8 = {};                             // extra group (clang-23 6-arg form), zero
    __builtin_amdgcn_tensor_load_to_lds(d0, d1, dz4, dz4, dz8, /*cpol=*/0);
}

__global__ __launch_bounds__(32)
void gru_wmma_tdm_kernel(const float* __restrict__ x,
                         const float* __restrict__ w_ih,
                         const float* __restrict__ w_hh,
                         const float* __restrict__ b_ih,
                         const float* __restrict__ b_hh,
                         const float* __restrict__ lin_w,
                         const float* __restrict__ lin_b,
                         float* __restrict__ out)
{
    __shared__ __align__(16) float smem[2 * kBUF_DW];   // 49,664 B, double buffer

    const int lane  = threadIdx.x;                      // single wave32 workgroup
    const int mbase = blockIdx.x * (kGRPS * kSEQ);
    // ISA 10.2: LDS flat address = {aperture, lds_offset} -> low 32 bits are the LDS byte address
    const unsigned lds0 = (unsigned)(uintptr_t)(void*)&smem[0];

    // Kick off DMA for tile 0; overlap descriptor-free work with it.
    tdm_load_tile(x + (size_t)mbase * kSEQ_DW, lds0);

    // ---- per-lane A fragments: A[g][k] = w_ih[g][4j+k], rows g>=3 zero ----
    // 16x4 f32 A layout: lanes 0-15 hold K=0,1 ; lanes 16-31 hold K=2,3
    const int row  = lane & 15;
    const int koff = (lane >> 4) * 2;
    v2f a[8];
    #pragma unroll
    for (int j = 0; j < 8; ++j) {
        float a0 = 0.0f, a1 = 0.0f;
        if (row < 3) {
            a0 = w_ih[row * kFE + 4 * j + koff];
            a1 = w_ih[row * kFE + 4 * j + koff + 1];
        }
        a[j][0] = a0; a[j][1] = a1;
    }

    v8f cinit = {};                       // C[g][n] seeded with b_ih[g]
    cinit[0] = b_ih[0]; cinit[1] = b_ih[1]; cinit[2] = b_ih[2];

    const float wh0 = w_hh[0], wh1 = w_hh[1], wh2 = w_hh[2];
    const float bh0 = b_hh[0], bh1 = b_hh[1], bh2 = b_hh[2];

    for (int g = 0; g < kGRPS; ++g) {
        const int buf = g & 1;
        if (g + 1 < kGRPS) {
            // Buffer (g+1)&1 was last read in the previous iteration; make sure all
            // outstanding LDS reads retired before the DMA overwrites it.
            asm volatile("s_wait_dscnt 0x0" ::: "memory");
            tdm_load_tile(x + (size_t)(mbase + (g + 1) * kSEQ) * kSEQ_DW,
                          lds0 + (unsigned)(((g + 1) & 1) * kBUF_DW * 4));
            __builtin_amdgcn_s_wait_tensorcnt(1);   // tile g resident (TDM in-order)
        } else {
            __builtin_amdgcn_s_wait_tensorcnt(0);   // last tile resident
        }

        const float* lbuf = &smem[buf * kBUF_DW + row * kROWSTR];
        float h = 0.0f;
        float hs[kT];

        #pragma unroll
        for (int t = 0; t < kT; ++t) {
            v8f c = cinit;
            #pragma unroll
            for (int j = 0; j < 8; ++j) {
                // 4x16 f32 B layout: lanes 0-15 hold K=0,1 ; lanes 16-31 hold K=2,3.
                // B[k][n] = x[m_n, f+k, t]; TDM pad -> +1 dw per 128 dw of payload.
                int f  = 4 * j + koff;
                int o0 = f * kT + t;
                int o1 = o0 + kT;
                v2f b;
                b[0] = lbuf[o0 + (o0 >> 7)];
                b[1] = lbuf[o1 + (o1 >> 7)];
                c = __builtin_amdgcn_wmma_f32_16x16x4_f32(
                        false, a[j], false, b, (short)0, c, false, false);
            }
            // lane n (0..15) holds gates g=0..2 of its sequence in c[0..2]
            float gr = c[0], gz = c[1], gn = c[2];
            float r_ = 1.0f / (1.0f + __expf(-(gr + wh0 * h + bh0)));
            float z_ = 1.0f / (1.0f + __expf(-(gz + wh1 * h + bh1)));
            float na = gn + r_ * (wh2 * h + bh2);
            float e2 = __expf(-2.0f * na);
            float n_ = (1.0f - e2) / (1.0f + e2);   // tanh(na)
            h = (1.0f - z_) * n_ + z_ * h;
            hs[t] = h;
        }

        // ---- linear head: out[m, o] = sum_t relu(h_t) * lin_w[o, t] + lin_b[o] ----
        if (lane < 16) {
            #pragma unroll
            for (int o = 0; o < 3; ++o) {
                float acc = lin_b[o];
                #pragma unroll
                for (int t = 0; t < kT; ++t)
                    acc += fmaxf(hs[t], 0.0f) * lin_w[o * kT + t];
                out[(size_t)(mbase + g * kSEQ + lane) * 3 + o] = acc;
            }
        }
    }
}

extern "C" void kernel_launch(void* const* d_in, const int* in_sizes, int n_in,
                              void* d_out, int out_size, void* d_ws, size_t ws_size,
                              hipStream_t stream) {
    const float* x     = (const float*)d_in[0];
    const float* w_ih  = (const float*)d_in[1];
    const float* w_hh  = (const float*)d_in[2];
    const float* b_ih  = (const float*)d_in[3];
    const float* b_hh  = (const float*)d_in[4];
    const float* lin_w = (const float*)d_in[5];
    const float* lin_b = (const float*)d_in[6];
    float* out = (float*)d_out;

    const int M = in_sizes[0] / (kFE * kT);       // 65536 sequences
    const int blocks = M / (kGRPS * kSEQ);        // 1024 single-wave workgroups
    gru_wmma_tdm_kernel<<<blocks, 32, 0, stream>>>(x, w_ih, w_hh, b_ih, b_hh,
                                                   lin_w, lin_b, out);
}